// nh_spa_mapper_simple_85873576116771
// MI455X (gfx1250) — compile-verified
//
#include <hip/hip_runtime.h>

// Problem constants from the reference setup.
#define B_   2
#define T_   8192
#define S_   8192
#define E_   256
#define NH_  16
#define MD_  16
#define LN_EPS 1e-5f

typedef __attribute__((ext_vector_type(2))) float v2f;
typedef __attribute__((ext_vector_type(8))) float v8f;
typedef unsigned long long u64;
typedef unsigned int       u32;

__device__ __forceinline__ void compiler_fence() { asm volatile("" ::: "memory"); }

// Monotonic float -> ordered u32 (handles the tiny-negative-from-FMA case).
__device__ __forceinline__ u32 f32_ordered(float f) {
  u32 b = __float_as_uint(f);
  return b ^ ((u32)((int)b >> 31) | 0x80000000u);
}

// ---------------------------------------------------------------------------
// Kernel A: brute-force KNN via V_WMMA_F32_16X16X4_F32.
// One wave32 handles 16 targets. Per 16-source tile:
//   A (16x4): lanes 0-15 hold (xt, yt) for target m=lane, K=2..3 rows zero.
//   B (4x16): lanes 0-15 hold (-2*xs, -2*ys) for source n=lane, K=2..3 zero.
//   C (16x16): ct2[m] + cs2[n]  ->  D = squared distance tile.
// D is dumped to LDS (wave-synchronous, DS in-order) and all 32 lanes scan:
// lane l keeps a sorted top-16 for target (l%16) over columns [8*(l/16), +8)
// as u64 keys (dist_bits<<32 | idx)  -> lexicographic (dist, idx) = top_k
// order.  Insert = replace max + one bubble pass of 15 compare-swaps, each a
// same-condition min/max select pair (v_cmp_lt_u64 + cndmask, no branches).
// ---------------------------------------------------------------------------
__global__ __launch_bounds__(256) void knn_kernel(const float* __restrict__ ct,
                                                  const float* __restrict__ cs,
                                                  int* __restrict__ knn) {
  __shared__ float dtile[8][256];      // per-wave 16x16 dist tile (8 KB)
  __shared__ u64   mK[8][32][16];      // per-wave merge scratch   (32 KB)

  const int w    = threadIdx.x >> 5;
  const int lane = threadIdx.x & 31;
  const int col  = lane & 15;
  const int h    = lane >> 4;

  const int gwave = blockIdx.x * 8 + w;     // 0 .. B_*T_/16 - 1
  const int bt0   = gwave * 16;             // first flattened target
  const int bidx  = bt0 / T_;
  const int t0    = bt0 - bidx * T_;        // 16 | T_, no batch straddle

  const float* ctb = ct + (size_t)bidx * 2 * T_;
  const float* csb = cs + (size_t)bidx * 2 * S_;

  // A operand (K rows 2,3 are zero -> lanes 16-31 carry zeros)
  v2f a;
  if (h == 0) { a.x = ctb[t0 + col]; a.y = ctb[T_ + t0 + col]; }
  else        { a.x = 0.f;           a.y = 0.f; }

  // Per-lane C contribution: ct2 for the 8 rows m = 8*h + r this lane holds.
  float ct2v[8];
#pragma unroll
  for (int r = 0; r < 8; ++r) {
    int m = h * 8 + r;
    float xx = ctb[t0 + m], yy = ctb[T_ + t0 + m];
    ct2v[r] = xx * xx + yy * yy;
  }

  // Sorted (ascending) top-16 keys.
  u64 l[16];
#pragma unroll
  for (int j = 0; j < 16; ++j) l[j] = ~0ull;

  float* dt = dtile[w];

  for (int tile = 0; tile < S_ / 16; ++tile) {
    const int n = tile * 16 + col;
    float sx = csb[n], sy = csb[S_ + n];
    v2f bv;
    if (h == 0) { bv.x = -2.f * sx; bv.y = -2.f * sy; }
    else        { bv.x = 0.f;       bv.y = 0.f; }
    float cs2 = sx * sx + sy * sy;

    v8f c;
#pragma unroll
    for (int r = 0; r < 8; ++r) c[r] = ct2v[r] + cs2;

    // D = A*B + C : squared distances for this 16x16 tile
    v8f d = __builtin_amdgcn_wmma_f32_16x16x4_f32(
        false, a, false, bv, (short)0, c, false, false);

    // C/D layout: lane l, vgpr r -> (m = r + 8*(l/16), n = l%16)
#pragma unroll
    for (int r = 0; r < 8; ++r) dt[lane * 8 + r] = d[r];
    compiler_fence();
    __builtin_amdgcn_wave_barrier();

    // Scan: lane l handles target m=col, local columns [8h, 8h+8).
#pragma unroll
    for (int q = 0; q < 8; ++q) {
      const int nloc = h * 8 + q;
      const float dd = dt[(((col >> 3) << 4) + nloc) * 8 + (col & 7)];
      const u64 key = ((u64)f32_ordered(dd) << 32) | (u32)(tile * 16 + nloc);
      if (key < l[15]) {
        l[15] = key;
#pragma unroll
        for (int j = 15; j >= 1; --j) {   // single bubble pass restores order
          u64 x0 = l[j - 1], x1 = l[j];
          bool sw = x1 < x0;              // same-cond min/max -> cndmask pair
          l[j - 1] = sw ? x1 : x0;
          l[j]     = sw ? x0 : x1;
        }
      }
    }
    __builtin_amdgcn_wave_barrier();
    compiler_fence();
  }

  // Merge the two per-target half-lists (lane l and l+16); key compare gives
  // (dist asc, idx asc) order directly.
#pragma unroll
  for (int j = 0; j < 16; ++j) mK[w][lane][j] = l[j];
  compiler_fence();
  __builtin_amdgcn_wave_barrier();

  if (h == 0) {
    int ia = 0, ib = 0;
    int* outp = knn + (size_t)(bt0 + col) * NH_;
    for (int j = 0; j < 16; ++j) {        // ia+ib == j <= 15, always in range
      u64 kA = mK[w][col][ia];
      u64 kB = mK[w][col + 16][ib];
      bool takeA = kA < kB;
      u64 kSel = takeA ? kA : kB;
      outp[j] = (int)(u32)kSel;
      ia += takeA ? 1 : 0;
      ib += takeA ? 0 : 1;
    }
  }
}

// ---------------------------------------------------------------------------
// Kernel B: PE -> LayerNorm -> k-proj -> softmax -> gathered weighted sum.
// One wave per target (8 targets / 256-thread block). Lanes 0-15 own one
// neighbor each (LN over 16 channels is lane-local). Wk staged in LDS once.
// Logits reduced across the wave with shfl_xor; gather phase streams 16
// contiguous 1 KB feature rows per target (2x b128 per lane, fully coalesced).
// ---------------------------------------------------------------------------
__global__ __launch_bounds__(256) void attn_kernel(
    const float* __restrict__ x,   const float* __restrict__ ct,
    const float* __restrict__ cs,  const float* __restrict__ Wpe,
    const float* __restrict__ bpe, const float* __restrict__ lnw,
    const float* __restrict__ lnb, const float* __restrict__ Wk,
    const float* __restrict__ bk,  const int* __restrict__ knn,
    float* __restrict__ out) {
  __shared__ float sWk[NH_ * MD_ * NH_];   // 4096 floats = 16 KB
  {
    const float4* src = (const float4*)Wk;
    float4*       dst = (float4*)sWk;
    for (int i = threadIdx.x; i < (NH_ * MD_ * NH_) / 4; i += 256) dst[i] = src[i];
  }
  __syncthreads();

  const int w    = threadIdx.x >> 5;
  const int lane = threadIdx.x & 31;
  const int n    = lane & 15;
  const int h    = lane >> 4;
  const int bt   = blockIdx.x * 8 + w;
  const int bidx = bt / T_;
  const int tt   = bt - bidx * T_;

  float p[16];
#pragma unroll
  for (int j = 0; j < 16; ++j) p[j] = 0.f;

  int myidx = 0;
  if (h == 0) {
    myidx = knn[(size_t)bt * NH_ + n];
    const float* ctb = ct + (size_t)bidx * 2 * T_;
    const float* csb = cs + (size_t)bidx * 2 * S_;
    float rx = csb[myidx]       - ctb[tt];
    float ry = csb[S_ + myidx]  - ctb[T_ + tt];

    float pe[16];
    float mu = 0.f;
#pragma unroll
    for (int c = 0; c < 16; ++c) {
      pe[c] = rx * Wpe[c] + ry * Wpe[16 + c] + bpe[c];
      mu += pe[c];
    }
    mu *= (1.f / 16.f);
    float var = 0.f;
#pragma unroll
    for (int c = 0; c < 16; ++c) { float dv = pe[c] - mu; var += dv * dv; }
    var *= (1.f / 16.f);
    float rstd = rsqrtf(var + LN_EPS);

#pragma unroll
    for (int c = 0; c < 16; ++c) {
      float kk = (pe[c] - mu) * rstd * lnw[c] + lnb[c];
      const float* row = &sWk[(n * 16 + c) * 16];
#pragma unroll
      for (int j = 0; j < 16; ++j) p[j] += kk * row[j];
    }
  }

  // Full wave32 reduction; lanes 16-31 contribute zeros.
#pragma unroll
  for (int mask = 16; mask >= 1; mask >>= 1) {
#pragma unroll
    for (int j = 0; j < 16; ++j) p[j] += __shfl_xor(p[j], mask, 32);
  }

  // Softmax over the 16 neighbor logits (identical in every lane).
#pragma unroll
  for (int j = 0; j < 16; ++j) p[j] += bk[j];
  float mx = p[0];
#pragma unroll
  for (int j = 1; j < 16; ++j) mx = fmaxf(mx, p[j]);
  float ssum = 0.f;
#pragma unroll
  for (int j = 0; j < 16; ++j) { p[j] = __expf(p[j] - mx); ssum += p[j]; }
  float inv = 1.f / ssum;
#pragma unroll
  for (int j = 0; j < 16; ++j) p[j] *= inv;

  // Attention-weighted neighbor-feature sum: lane covers features lane*8..+7.
  float acc[8];
#pragma unroll
  for (int q = 0; q < 8; ++q) acc[q] = 0.f;
  const float* xb = x + (size_t)bidx * S_ * E_;
#pragma unroll
  for (int j = 0; j < 16; ++j) {
    int sj = __shfl(myidx, j, 32);
    const float* row = xb + (size_t)sj * E_ + lane * 8;
    float4 a0 = *(const float4*)(row);
    float4 a1 = *(const float4*)(row + 4);
    float wg = p[j];
    acc[0] += wg * a0.x; acc[1] += wg * a0.y; acc[2] += wg * a0.z; acc[3] += wg * a0.w;
    acc[4] += wg * a1.x; acc[5] += wg * a1.y; acc[6] += wg * a1.z; acc[7] += wg * a1.w;
  }
  float* op = out + (size_t)bt * E_ + lane * 8;
  float4 o0 = {acc[0], acc[1], acc[2], acc[3]};
  float4 o1 = {acc[4], acc[5], acc[6], acc[7]};
  *(float4*)op       = o0;
  *(float4*)(op + 4) = o1;
}

extern "C" void kernel_launch(void* const* d_in, const int* in_sizes, int n_in,
                              void* d_out, int out_size, void* d_ws, size_t ws_size,
                              hipStream_t stream) {
  (void)in_sizes; (void)n_in; (void)out_size; (void)ws_size;
  const float* x   = (const float*)d_in[0];
  const float* ct  = (const float*)d_in[1];
  const float* cs  = (const float*)d_in[2];
  const float* Wpe = (const float*)d_in[3];
  const float* bpe = (const float*)d_in[4];
  const float* lnw = (const float*)d_in[5];
  const float* lnb = (const float*)d_in[6];
  const float* Wk  = (const float*)d_in[7];
  const float* bk  = (const float*)d_in[8];
  float* outp = (float*)d_out;
  int*   knn  = (int*)d_ws;   // B_*T_*NH_ int32 = 1 MB scratch

  // 1024 waves of 16 targets each -> 128 blocks x 8 waves.
  knn_kernel<<<(B_ * T_) / (16 * 8), 256, 0, stream>>>(ct, cs, knn);
  // 1 wave per target -> 2048 blocks x 8 waves.
  attn_kernel<<<(B_ * T_) / 8, 256, 0, stream>>>(x, ct, cs, Wpe, bpe, lnw, lnb,
                                                 Wk, bk, knn, outp);
}